// HybridMPCController_80650895884498
// MI455X (gfx1250) — compile-verified
//
#include <hip/hip_runtime.h>
#include <math.h>

#define ALPHA_P 0.5f
#define LRC     0.05f

typedef float v2f __attribute__((ext_vector_type(2)));
typedef float v8f __attribute__((ext_vector_type(8)));

// ---------------------------------------------------------------- utilities
__global__ void k_zero(float* __restrict__ p, int n) {
    int i = blockIdx.x * blockDim.x + threadIdx.x;
    if (i < n) p[i] = 0.f;
}

// ---------------------------------------------------------------- z1 = W1 @ [xa;xb]  (accumulate via atomics)
// grid = (2N/4096, 8), block = 256. Each block: 16 rows x 4096 cols partial.
__global__ void k_w1_matvec(const float* __restrict__ W1,
                            const float* __restrict__ xa,
                            const float* __restrict__ xb,
                            float* __restrict__ z1, int N) {
    const size_t N2 = (size_t)N * 2;
    const int r0 = blockIdx.y * 16;
    const int c0 = blockIdx.x * 4096;
    float acc[16];
#pragma unroll
    for (int r = 0; r < 16; ++r) acc[r] = 0.f;
    for (int c = c0 + (int)threadIdx.x; c < c0 + 4096; c += 256) {
        float xv = (c < N) ? xa[c] : xb[c - N];
        const float* wp = W1 + c;
#pragma unroll
        for (int r = 0; r < 16; ++r) acc[r] += wp[(size_t)(r0 + r) * N2] * xv;
    }
    __shared__ float red[256];
#pragma unroll 1
    for (int r = 0; r < 16; ++r) {
        red[threadIdx.x] = acc[r];
        __syncthreads();
        for (int s = 128; s >= 1; s >>= 1) {
            if ((int)threadIdx.x < s) red[threadIdx.x] += red[threadIdx.x + s];
            __syncthreads();
        }
        if (threadIdx.x == 0) atomicAdd(&z1[r0 + r], red[0]);
        __syncthreads();
    }
}

// ---------------------------------------------------------------- h1 = tanh(z1+b1); h2 = tanh(W2@h1+b2). 1 block, 128 thr
__global__ void k_mlp_mid(const float* __restrict__ z1, const float* __restrict__ b1,
                          const float* __restrict__ W2, const float* __restrict__ b2,
                          float* __restrict__ h1o, float* __restrict__ h2o) {
    __shared__ float h1[128];
    int r = threadIdx.x;
    float v = tanhf(z1[r] + b1[r]);
    h1[r] = v; h1o[r] = v;
    __syncthreads();
    float acc = b2[r];
    const float* w = W2 + r * 128;
#pragma unroll 8
    for (int c = 0; c < 128; ++c) acc += w[c] * h1[c];
    h2o[r] = tanhf(acc);
}

// ---------------------------------------------------------------- m = W3 @ h2 + b3  via V_WMMA_F32_16X16X4_F32
// block = 128 (4 waves), each block handles 64 rows; grid = N/64.
// Staging: CDNA5 async-to-LDS copies (ASYNCcnt), then per-wave s_wait_asynccnt + barrier.
// A-tile (f32 16x4): lanes0-15 M=0..15, vgpr0={K0|K2}, vgpr1={K1|K3} -> per-lane v2f from LDS tile.
// B (4x16): all 16 columns identical = h2 chunk -> every column of D holds the matvec partials.
__global__ void k_w3_fwd_wmma(const float* __restrict__ W3, const float* __restrict__ h2,
                              const float* __restrict__ b3, float* __restrict__ m, int N) {
    __shared__ float tile[64 * 132];   // padded stride (132) to dodge LDS bank conflicts; rows stay 16B-aligned (528B)
    __shared__ float hs[128];
    const int tid = threadIdx.x;
    const int rowBase = blockIdx.x * 64;
    if (tid < 128) hs[tid] = h2[tid];
    // async copy global->LDS: 64 rows x 32 float4 per block, one B128 per lane per iter
    for (int i = tid; i < 64 * 32; i += 128) {
        int r = i >> 5, c4 = i & 31;
        unsigned lds_dst = (unsigned)(uintptr_t)(&tile[r * 132 + c4 * 4]);
        const float* gsrc = W3 + (size_t)(rowBase + r) * 128 + c4 * 4;
        asm volatile("global_load_async_to_lds_b128 %0, %1, off"
                     :: "v"(lds_dst), "v"(gsrc)
                     : "memory");
    }
    asm volatile("s_wait_asynccnt 0x0" ::: "memory");
    __syncthreads();
    const int wave = tid >> 5;
    const int lane = tid & 31;
    const int wr   = wave * 16;
    const int rl   = wr + (lane & 15);
    const int ksel = (lane >> 4) << 1;   // 0 for lanes 0-15, 2 for lanes 16-31
    v8f acc = {};
#pragma unroll
    for (int k = 0; k < 128; k += 4) {
        int kk = k + ksel;
        v2f a; a.x = tile[rl * 132 + kk]; a.y = tile[rl * 132 + kk + 1];
        v2f b; b.x = hs[kk];              b.y = hs[kk + 1];
        acc = __builtin_amdgcn_wmma_f32_16x16x4_f32(false, a, false, b, (short)0, acc, false, false);
    }
    // D layout: vgpr r -> M=r (lanes0-15, col N=0 at lane0), M=8+r (lanes16-31, col N=0 at lane16)
    if (lane == 0 || lane == 16) {
        int ro = (lane == 16) ? 8 : 0;
#pragma unroll
        for (int r = 0; r < 8; ++r) {
            int row = rowBase + wr + ro + r;
            m[row] = acc[r] + b3[row];
        }
    }
}

// ---------------------------------------------------------------- forward plant step + sum(y) reduction
__global__ void k_fwd_step(const float* __restrict__ m, const float* __restrict__ St,
                           const float* __restrict__ yp, const float* __restrict__ kg,
                           const float* __restrict__ lo, const float* __restrict__ hi,
                           float* __restrict__ Yt, float* __restrict__ tot, int N) {
    int i = blockIdx.x * 256 + threadIdx.x;
    float l = lo[i], h = hi[i];
    float span = fmaxf(h - l, 1e-6f);
    float sg = 1.f / (1.f + expf(-m[i]));
    float p  = l + span * sg;
    float s  = St[i];
    float cp = s + 0.1f * (p - s);
    float corr = fminf(fmaxf(cp, l), h);
    float yn = (1.f - ALPHA_P) * yp[i] + ALPHA_P * kg[i] * corr;
    Yt[i] = yn;
    float v = yn;
#pragma unroll
    for (int o = 16; o > 0; o >>= 1) v += __shfl_down(v, o, 32);
    __shared__ float wsum[8];
    int lane = threadIdx.x & 31, wv = threadIdx.x >> 5;
    if (lane == 0) wsum[wv] = v;
    __syncthreads();
    if (threadIdx.x == 0) {
        float t = 0.f;
#pragma unroll
        for (int w = 0; w < 8; ++w) t += wsum[w];
        atomicAdd(tot, t);
    }
}

// ---------------------------------------------------------------- Sdot = sum((y/T - tr) * y/T)
__global__ void k_comp_dot(const float* __restrict__ Yt, const float* __restrict__ tr,
                           const float* __restrict__ tot, float* __restrict__ sdot, int N) {
    int i = blockIdx.x * 256 + threadIdx.x;
    float T = tot[0] + 1e-12f;
    float comp = Yt[i] / T;
    float v = (comp - tr[i]) * comp;
#pragma unroll
    for (int o = 16; o > 0; o >>= 1) v += __shfl_down(v, o, 32);
    __shared__ float wsum[8];
    int lane = threadIdx.x & 31, wv = threadIdx.x >> 5;
    if (lane == 0) wsum[wv] = v;
    __syncthreads();
    if (threadIdx.x == 0) {
        float t = 0.f;
#pragma unroll
        for (int w = 0; w < 8; ++w) t += wsum[w];
        atomicAdd(sdot, t);
    }
}

// ---------------------------------------------------------------- lam = dL/dy for final stage
__global__ void k_stage_grad(const float* __restrict__ Yt, const float* __restrict__ tr,
                             const float* __restrict__ sc2, const float* __restrict__ Ft,
                             float* __restrict__ lam, int N) {
    int i = blockIdx.x * 256 + threadIdx.x;
    float T = sc2[0] + 1e-12f, Sd = sc2[1];
    lam[i] = (6.f / T) * (Yt[i] / T - tr[i] - Sd) + 12.f * (T - Ft[0]);
}

// ---------------------------------------------------------------- backward: local grads through clip/sigmoid, produce g_m and direct S grad
__global__ void k_bwd_stage1(const float* __restrict__ lam, const float* __restrict__ m,
                             const float* __restrict__ St, const float* __restrict__ kg,
                             const float* __restrict__ lo, const float* __restrict__ hi,
                             float* __restrict__ gm, float* __restrict__ Gt, int N) {
    int i = blockIdx.x * 256 + threadIdx.x;
    float l = lo[i], h = hi[i];
    float span = fmaxf(h - l, 1e-6f);
    float sg = 1.f / (1.f + expf(-m[i]));
    float p  = l + span * sg;
    float s  = St[i];
    float cp = s + 0.1f * (p - s);
    float u  = ALPHA_P * kg[i] * lam[i];
    float gcp = (cp > l && cp < h) ? u : 0.f;   // clip pullback
    Gt[i] = 0.9f * gcp;                          // direct d corr / d s
    gm[i] = 0.1f * gcp * span * sg * (1.f - sg); // through map_speed
}

// ---------------------------------------------------------------- gh2 = W3^T @ gm. block=128 (one thread/col), grid=N/256
__global__ void k_w3t(const float* __restrict__ W3, const float* __restrict__ gm,
                      float* __restrict__ gh2, int N) {
    int c = threadIdx.x;
    int r0 = blockIdx.x * 256;
    float acc = 0.f;
    for (int r = 0; r < 256; ++r) acc += W3[(size_t)(r0 + r) * 128 + c] * gm[r0 + r];
    atomicAdd(&gh2[c], acc);
}

// ---------------------------------------------------------------- gz1 = (W2^T (gh2*(1-h2^2))) * (1-h1^2). 1 block, 128 thr
__global__ void k_mid_bwd(const float* __restrict__ gh2, const float* __restrict__ h1,
                          const float* __restrict__ h2, const float* __restrict__ W2,
                          float* __restrict__ gz1) {
    __shared__ float gz2[128];
    int r = threadIdx.x;
    float h2v = h2[r];
    gz2[r] = gh2[r] * (1.f - h2v * h2v);
    __syncthreads();
    float acc = 0.f;
#pragma unroll 8
    for (int q = 0; q < 128; ++q) acc += W2[q * 128 + r] * gz2[q];
    float h1v = h1[r];
    gz1[r] = acc * (1.f - h1v * h1v);
}

// ---------------------------------------------------------------- gx = W1^T @ gz1, split into (y-half, s-half). grid=2N/256
__global__ void k_w1t(const float* __restrict__ W1, const float* __restrict__ gz1,
                      float* __restrict__ gxy, float* __restrict__ gxs, int N) {
    __shared__ float g[128];
    if (threadIdx.x < 128) g[threadIdx.x] = gz1[threadIdx.x];
    __syncthreads();
    int c = blockIdx.x * 256 + threadIdx.x;
    const size_t N2 = (size_t)N * 2;
    float acc = 0.f;
#pragma unroll 8
    for (int r = 0; r < 128; ++r) acc += W1[(size_t)r * N2 + c] * g[r];
    if (c < N) gxy[c] = acc; else gxs[c - N] = acc;
}

// ---------------------------------------------------------------- combine: G[t]+=gxs; lam = (1-a)lam + gxy + stageGrad(t-1)
__global__ void k_bwd_combine(float* __restrict__ Gt, const float* __restrict__ gxs,
                              const float* __restrict__ gxy, float* __restrict__ lam,
                              const float* __restrict__ Yp, const float* __restrict__ tr,
                              const float* __restrict__ sc2, const float* __restrict__ Ft,
                              int hasPrev, int N) {
    int i = blockIdx.x * 256 + threadIdx.x;
    Gt[i] += gxs[i];
    if (hasPrev) {
        float T = sc2[0] + 1e-12f, Sd = sc2[1];
        float sgr = (6.f / T) * (Yp[i] / T - tr[i] - Sd) + 12.f * (T - Ft[0]);
        lam[i] = (1.f - ALPHA_P) * lam[i] + gxy[i] + sgr;
    }
}

// ---------------------------------------------------------------- SGD step: smoothness + barrier grads, update, clamp.
// one thread per column i, loops over all 5 time rows (avoids cross-t races)
__global__ void k_sgd_update(float* __restrict__ S, const float* __restrict__ G,
                             const float* __restrict__ lo, const float* __restrict__ hi, int N) {
    int i = blockIdx.x * 256 + threadIdx.x;
    float l = lo[i], h = hi[i];
    float span = fmaxf(h - l, 1e-6f);
    float s[5], g[5];
#pragma unroll
    for (int t = 0; t < 5; ++t) { s[t] = S[(size_t)t * N + i]; g[t] = G[(size_t)t * N + i]; }
#pragma unroll
    for (int t = 0; t < 5; ++t) {
        float gt = g[t];
        if (t > 0) gt += 0.1f * (s[t] - s[t - 1]);
        if (t < 4) gt -= 0.1f * (s[t + 1] - s[t]);
        float tau = (s[t] - l) / span;
        float bg = 0.f;
        if (tau > 0.02f)       bg += 1.f / tau;
        if (1.f - tau > 0.02f) bg -= 1.f / (1.f - tau);
        gt += -0.08f * bg / span;
        float sn = s[t] - LRC * gt;
        S[(size_t)t * N + i] = fminf(fmaxf(sn, l), h);
    }
}

// ---------------------------------------------------------------- warm start: S[t] = clip(prev + 0.2*(map(m)-prev))
__global__ void k_warm_init(const float* __restrict__ m, const float* __restrict__ prev,
                            const float* __restrict__ lo, const float* __restrict__ hi,
                            float* __restrict__ S, int N) {
    int i = blockIdx.x * 256 + threadIdx.x;
    float l = lo[i], h = hi[i];
    float span = fmaxf(h - l, 1e-6f);
    float sg = 1.f / (1.f + expf(-m[i]));
    float bias = l + span * sg;
    float s0 = fminf(fmaxf(prev[i] + 0.2f * (bias - prev[i]), l), h);
#pragma unroll
    for (int t = 0; t < 5; ++t) S[(size_t)t * N + i] = s0;
}

// ---------------------------------------------------------------- final slew limit
__global__ void k_final(const float* __restrict__ S0, const float* __restrict__ prev,
                        const float* __restrict__ lo, const float* __restrict__ hi,
                        float* __restrict__ out, int N) {
    int i = blockIdx.x * 256 + threadIdx.x;
    float d = S0[i] - prev[i];
    d = fminf(fmaxf(d, -2.f), 2.f);
    out[i] = fminf(fmaxf(prev[i] + d, lo[i]), hi[i]);
}

// ================================================================ host
static void mlp_forward(const float* W1, const float* b1, const float* W2, const float* b2,
                        const float* W3, const float* b3,
                        const float* xa, const float* xb,
                        float* z1, float* h1o, float* h2o, float* mout,
                        int N, hipStream_t stream) {
    k_zero<<<1, 128, 0, stream>>>(z1, 128);
    dim3 g1((2 * N) / 4096, 8);
    k_w1_matvec<<<g1, 256, 0, stream>>>(W1, xa, xb, z1, N);
    k_mlp_mid<<<1, 128, 0, stream>>>(z1, b1, W2, b2, h1o, h2o);
    k_w3_fwd_wmma<<<N / 64, 128, 0, stream>>>(W3, h2o, b3, mout, N);
}

extern "C" void kernel_launch(void* const* d_in, const int* in_sizes, int n_in,
                              void* d_out, int out_size, void* d_ws, size_t ws_size,
                              hipStream_t stream) {
    (void)n_in; (void)out_size; (void)ws_size;
    const float* y    = (const float*)d_in[0];
    const float* tr   = (const float*)d_in[1];
    const float* Ft   = (const float*)d_in[2];
    const float* prev = (const float*)d_in[3];
    const float* kg   = (const float*)d_in[4];
    const float* lo   = (const float*)d_in[5];
    const float* hi   = (const float*)d_in[6];
    const float* W1   = (const float*)d_in[7];
    const float* b1   = (const float*)d_in[8];
    const float* W2   = (const float*)d_in[9];
    const float* b2   = (const float*)d_in[10];
    const float* W3   = (const float*)d_in[11];
    const float* b3   = (const float*)d_in[12];
    float* out = (float*)d_out;
    const int N = in_sizes[0];
    const size_t Ns = (size_t)N;

    float* ws  = (float*)d_ws;
    float* S   = ws;            // 5N shooting vars
    float* G   = S   + 5 * Ns;  // 5N MLP/direct grad
    float* Y   = G   + 5 * Ns;  // 5N y trajectory
    float* M   = Y   + 5 * Ns;  // 5N mlp outputs per t
    float* lam = M   + 5 * Ns;  // N adjoint of y_t
    float* gm  = lam + Ns;      // N (also warm-start m scratch)
    float* gxy = gm  + Ns;      // N
    float* gxs = gxy + Ns;      // N
    float* z1  = gxs + Ns;      // 128
    float* gz1 = z1  + 128;     // 128
    float* gh2 = gz1 + 128;     // 128
    float* h1s = gh2 + 128;     // 5*128
    float* h2s = h1s + 640;     // 5*128
    float* scal= h2s + 640;     // [t*2]=tot, [t*2+1]=Sdot

    const int egrid = N / 256;

    // warm start
    mlp_forward(W1, b1, W2, b2, W3, b3, y, prev, z1, h1s, h2s, gm, N, stream);
    k_warm_init<<<egrid, 256, 0, stream>>>(gm, prev, lo, hi, S, N);

    for (int it = 0; it < 8; ++it) {
        // ---------------- forward rollout ----------------
        for (int t = 0; t < 5; ++t) {
            const float* yp = (t == 0) ? y : (Y + (size_t)(t - 1) * Ns);
            mlp_forward(W1, b1, W2, b2, W3, b3, yp, S + (size_t)t * Ns,
                        z1, h1s + t * 128, h2s + t * 128, M + (size_t)t * Ns, N, stream);
            k_zero<<<1, 32, 0, stream>>>(scal + 2 * t, 2);
            k_fwd_step<<<egrid, 256, 0, stream>>>(M + (size_t)t * Ns, S + (size_t)t * Ns, yp,
                                                  kg, lo, hi, Y + (size_t)t * Ns, scal + 2 * t, N);
            k_comp_dot<<<egrid, 256, 0, stream>>>(Y + (size_t)t * Ns, tr, scal + 2 * t,
                                                  scal + 2 * t + 1, N);
        }
        // ---------------- backward (adjoint) ----------------
        k_stage_grad<<<egrid, 256, 0, stream>>>(Y + 4 * Ns, tr, scal + 8, Ft, lam, N);
        for (int t = 4; t >= 0; --t) {
            k_bwd_stage1<<<egrid, 256, 0, stream>>>(lam, M + (size_t)t * Ns, S + (size_t)t * Ns,
                                                    kg, lo, hi, gm, G + (size_t)t * Ns, N);
            k_zero<<<1, 128, 0, stream>>>(gh2, 128);
            k_w3t<<<N / 256, 128, 0, stream>>>(W3, gm, gh2, N);
            k_mid_bwd<<<1, 128, 0, stream>>>(gh2, h1s + t * 128, h2s + t * 128, W2, gz1);
            k_w1t<<<(2 * N) / 256, 256, 0, stream>>>(W1, gz1, gxy, gxs, N);
            const float* yp = (t > 0) ? (Y + (size_t)(t - 1) * Ns) : y;
            const float* sc = (t > 0) ? (scal + 2 * (t - 1)) : scal;
            k_bwd_combine<<<egrid, 256, 0, stream>>>(G + (size_t)t * Ns, gxs, gxy, lam,
                                                     yp, tr, sc, Ft, (t > 0) ? 1 : 0, N);
        }
        k_sgd_update<<<egrid, 256, 0, stream>>>(S, G, lo, hi, N);
    }
    k_final<<<egrid, 256, 0, stream>>>(S, prev, lo, hi, out, N);
}